// AxialAttention_9371618640634
// MI455X (gfx1250) — compile-verified
//
#include <hip/hip_runtime.h>
#include <hip/hip_bf16.h>
#include <math.h>

typedef float v2f __attribute__((ext_vector_type(2)));
typedef float v4f __attribute__((ext_vector_type(4)));
typedef float v8f __attribute__((ext_vector_type(8)));

// Geometry: q/k/v are (b=4, heads=8, t=16, h=32, w=32, d=64), attention along h.
#define LAX 32                 // sequence length (h axis)
#define WW  32                 // w axis
#define DD  64                 // head dim
#define ROW_STRIDE (WW*DD)     // 2048 floats between consecutive h rows
#define SLAB (LAX*WW*DD)       // 65536 floats per (b,heads,t)
#define NPROB (4*8*16*32)      // 16384 attention problems

#define WAVES 4                // problems (waves) per workgroup
#define QK_PITCH 68            // dwords per LDS row for Q/K/V tiles (64 + 4 pad)
#define P_PITCH 36             // dwords per LDS row for the 32x32 prob matrix
#define TBUF (LAX*QK_PITCH)    // 2176 dwords per tensor tile
#define PROB_DW (3*TBUF)       // 6528 dwords per problem (Q,K,V); P reuses Q

static __device__ __forceinline__ v8f wmma_f32x4(v2f a, v2f b, v8f c) {
    // D = A(16x4 f32) * B(4x16 f32) + C(16x16 f32)
    return __builtin_amdgcn_wmma_f32_16x16x4_f32(
        false, a, false, b, (short)0, c, false, false);
}

__global__ void __launch_bounds__(WAVES*32)
axial_attn_kernel(const float* __restrict__ q,
                  const float* __restrict__ k,
                  const float* __restrict__ v,
                  float* __restrict__ out) {
    __shared__ float lds[WAVES * PROB_DW];   // ~102 KB

    const int tid  = threadIdx.x;
    const int wave = tid >> 5;
    const int lane = tid & 31;
    const int lr   = lane & 15;              // lane % 16
    const int hb   = lane >> 4;              // half-wave select

    const int p0   = blockIdx.x * WAVES;     // first problem of this block
    const int bht  = p0 / WW;                // (b,heads,t) slab index
    const int w0   = p0 % WW;                // WAVES consecutive w's share a slab
    const size_t slab4 = (size_t)bht * (SLAB / 4);   // slab offset in float4s

    // ---- Phase 0: cooperative, fully-coalesced global->LDS staging ----------
    // Per tensor: 4 problems x 32 rows x 64 floats = 2048 float4s, 16 rounds.
    const float* srcs[3] = { q, k, v };
#pragma unroll
    for (int tsel = 0; tsel < 3; ++tsel) {
        const v4f* src = (const v4f*)srcs[tsel];
#pragma unroll
        for (int i = 0; i < 16; ++i) {
            int idx = i * 128 + tid;         // 0..2047
            int row = idx >> 6;              // h row 0..31
            int q4  = idx & 63;              // float4 within the 1KB row chunk
            int j   = q4 >> 4;               // which problem (w0+j)
            int c4  = q4 & 15;               // float4 within the 64-float row
            v4f val = src[slab4 + (size_t)row * (ROW_STRIDE / 4)
                          + (size_t)(w0 + j) * (DD / 4) + c4];
            *(v4f*)&lds[j * PROB_DW + tsel * TBUF + row * QK_PITCH + c4 * 4] = val;
        }
    }
    __syncthreads();

    float* sQ = &lds[wave * PROB_DW + 0 * TBUF];
    float* sK = &lds[wave * PROB_DW + 1 * TBUF];
    float* sV = &lds[wave * PROB_DW + 2 * TBUF];
    float* sP = sQ;                          // Q is dead after phase 1; reuse

    // ---- Phase 1: S = Q * K^T  (32x32 = 2x2 tiles of 16x16, K-dim 64) -------
    v8f c00 = {}, c01 = {}, c10 = {}, c11 = {};
#pragma unroll
    for (int kt = 0; kt < 16; ++kt) {
        // A fragment: lane<16 -> (M=lr, K=kt*4+{0,1}); lane>=16 -> K=kt*4+{2,3}
        v2f aq0 = *(const v2f*)&sQ[lr * QK_PITCH + kt * 4 + hb * 2];
        v2f aq1 = *(const v2f*)&sQ[(lr + 16) * QK_PITCH + kt * 4 + hb * 2];
        // B fragment of K^T == rows of K: identical addressing pattern
        v2f bk0 = *(const v2f*)&sK[lr * QK_PITCH + kt * 4 + hb * 2];
        v2f bk1 = *(const v2f*)&sK[(lr + 16) * QK_PITCH + kt * 4 + hb * 2];
        c00 = wmma_f32x4(aq0, bk0, c00);
        c01 = wmma_f32x4(aq0, bk1, c01);
        c10 = wmma_f32x4(aq1, bk0, c10);
        c11 = wmma_f32x4(aq1, bk1, c11);
    }

    // ---- Phase 2: spill scaled S, per-row softmax, write P back -------------
    const float scale = 0.125f;              // 1/sqrt(64)
#pragma unroll
    for (int r = 0; r < 8; ++r) {
        int rowA = r + 8 * hb;               // C/D layout: VGPR r holds M=r / M=r+8
        sP[rowA * P_PITCH + lr]            = c00[r] * scale;
        sP[rowA * P_PITCH + 16 + lr]       = c01[r] * scale;
        sP[(16 + rowA) * P_PITCH + lr]     = c10[r] * scale;
        sP[(16 + rowA) * P_PITCH + 16 + lr]= c11[r] * scale;
    }
    __syncthreads();

    {   // lane == row of this wave's 32x32 score matrix
        float p[32];
#pragma unroll
        for (int cc = 0; cc < 8; ++cc) {
            v4f t = *(const v4f*)&sP[lane * P_PITCH + cc * 4];
            p[cc*4+0] = t[0]; p[cc*4+1] = t[1]; p[cc*4+2] = t[2]; p[cc*4+3] = t[3];
        }
        float m = p[0];
#pragma unroll
        for (int i = 1; i < 32; ++i) m = fmaxf(m, p[i]);
        float s = 0.f;
#pragma unroll
        for (int i = 0; i < 32; ++i) { p[i] = __expf(p[i] - m); s += p[i]; }
        float inv = 1.0f / s;
#pragma unroll
        for (int cc = 0; cc < 8; ++cc) {
            v4f t;
            t[0] = p[cc*4+0] * inv; t[1] = p[cc*4+1] * inv;
            t[2] = p[cc*4+2] * inv; t[3] = p[cc*4+3] * inv;
            *(v4f*)&sP[lane * P_PITCH + cc * 4] = t;
        }
    }
    __syncthreads();

    // ---- Phase 3: O = P * V  (32x64 = 2x4 tiles, K-dim 32) ------------------
    v8f o[2][4];
#pragma unroll
    for (int mt = 0; mt < 2; ++mt)
#pragma unroll
        for (int nt = 0; nt < 4; ++nt) o[mt][nt] = (v8f){};

#pragma unroll
    for (int kt = 0; kt < 8; ++kt) {
        v2f a0 = *(const v2f*)&sP[lr * P_PITCH + kt * 4 + hb * 2];
        v2f a1 = *(const v2f*)&sP[(lr + 16) * P_PITCH + kt * 4 + hb * 2];
#pragma unroll
        for (int nt = 0; nt < 4; ++nt) {
            // B fragment of V (dense, row-major): lane<16 rows kt*4+{0,1},
            // lane>=16 rows kt*4+{2,3}, column nt*16+lr
            v2f bv;
            bv[0] = sV[(kt * 4 + hb * 2 + 0) * QK_PITCH + nt * 16 + lr];
            bv[1] = sV[(kt * 4 + hb * 2 + 1) * QK_PITCH + nt * 16 + lr];
            o[0][nt] = wmma_f32x4(a0, bv, o[0][nt]);
            o[1][nt] = wmma_f32x4(a1, bv, o[1][nt]);
        }
    }

    // ---- Phase 4: store O from C/D-layout registers back to (…,h,w,d) -------
    float* outp = out + (size_t)bht * SLAB + (size_t)(w0 + wave) * DD;
#pragma unroll
    for (int mt = 0; mt < 2; ++mt)
#pragma unroll
        for (int nt = 0; nt < 4; ++nt)
#pragma unroll
            for (int r = 0; r < 8; ++r) {
                int row = mt * 16 + r + 8 * hb;
                int col = nt * 16 + lr;
                outp[(size_t)row * ROW_STRIDE + col] = o[mt][nt][r];
            }
}

extern "C" void kernel_launch(void* const* d_in, const int* in_sizes, int n_in,
                              void* d_out, int out_size, void* d_ws, size_t ws_size,
                              hipStream_t stream) {
    const float* q = (const float*)d_in[0];
    const float* k = (const float*)d_in[1];
    const float* v = (const float*)d_in[2];
    // d_in[3]=decode_step, d_in[4]=decode_idx: unused (decode_step==0 path)
    float* out = (float*)d_out;

    dim3 grid(NPROB / WAVES);   // 4096 workgroups, 4 problems each
    dim3 block(WAVES * 32);     // 128 threads = 4 wave32
    axial_attn_kernel<<<grid, block, 0, stream>>>(q, k, v, out);
}